// DWTDetailLoss_87986700026275
// MI455X (gfx1250) — compile-verified
//
#include <hip/hip_runtime.h>
#include <hip/hip_bf16.h>

// DWT Haar detail loss for (32,3,512,512) f32 tensors on MI455X (gfx1250).
// Bandwidth-bound (~402 MB total traffic -> ~17us at 23.3 TB/s).
// Pass1: global mins (for the (x+1)*0.5 affine flags). Pass2: fused 2-level
// Haar detail L1 sums, streaming tiles through LDS with gfx1250 async
// global->LDS copies (double buffered). Pass3: WMMA-based deterministic
// final reduction of per-block partials.

typedef float fvec4 __attribute__((ext_vector_type(4)));
typedef float v2f   __attribute__((ext_vector_type(2)));
typedef float v8f   __attribute__((ext_vector_type(8)));

#define IMGS            96          // 32*3
#define N4TOT           6291456     // float4 elements per tensor (96*512*128)
#define NB1             1024        // blocks for min pass
#define NB2             768         // blocks for dwt pass
#define TILES_PER_BLOCK 8           // 6144 tiles / 768 blocks

// ---------------------------------------------------------------- pass 1: min
__global__ __launch_bounds__(256)
void min_pass(const float* __restrict__ o, const float* __restrict__ t,
              float* __restrict__ minO, float* __restrict__ minT) {
  __shared__ float ro[256], rt[256];
  const int tid = threadIdx.x;
  const fvec4* o4 = (const fvec4*)o;
  const fvec4* t4 = (const fvec4*)t;
  float vo = 3.4e38f, vt = 3.4e38f;
  for (int i = blockIdx.x * 256 + tid; i < N4TOT; i += NB1 * 256) {
    fvec4 a = o4[i];
    fvec4 b = t4[i];
    vo = fminf(vo, fminf(fminf(a.x, a.y), fminf(a.z, a.w)));
    vt = fminf(vt, fminf(fminf(b.x, b.y), fminf(b.z, b.w)));
  }
  ro[tid] = vo; rt[tid] = vt;
  __syncthreads();
  for (int s = 128; s > 0; s >>= 1) {
    if (tid < s) {
      ro[tid] = fminf(ro[tid], ro[tid + s]);
      rt[tid] = fminf(rt[tid], rt[tid + s]);
    }
    __syncthreads();
  }
  if (tid == 0) { minO[blockIdx.x] = ro[0]; minT[blockIdx.x] = rt[0]; }
}

// ------------------------------------------------------------ pass 1b: flags
__global__ __launch_bounds__(256)
void flags_pass(const float* __restrict__ minO, const float* __restrict__ minT,
                float* __restrict__ flags) {
  __shared__ float ro[256], rt[256];
  const int tid = threadIdx.x;
  float vo = 3.4e38f, vt = 3.4e38f;
  for (int i = tid; i < NB1; i += 256) {
    vo = fminf(vo, minO[i]);
    vt = fminf(vt, minT[i]);
  }
  ro[tid] = vo; rt[tid] = vt;
  __syncthreads();
  for (int s = 128; s > 0; s >>= 1) {
    if (tid < s) {
      ro[tid] = fminf(ro[tid], ro[tid + s]);
      rt[tid] = fminf(rt[tid], rt[tid + s]);
    }
    __syncthreads();
  }
  if (tid == 0) {
    const bool no = ro[0] < 0.0f, nt = rt[0] < 0.0f;
    flags[0] = no ? 0.5f : 1.0f;   // scale  for output
    flags[1] = no ? 0.5f : 0.0f;   // offset for output
    flags[2] = nt ? 0.5f : 1.0f;   // scale  for target
    flags[3] = nt ? 0.5f : 0.0f;   // offset for target
  }
}

// -------------------------------------------------------------- pass 2: DWT
// Tile = 8 rows x 512 cols of both tensors (32 KB), double buffered in LDS.
// Data movement uses gfx1250 async global->LDS b128 copies (ASYNCcnt).
__global__ __launch_bounds__(256)
void dwt_pass(const float* __restrict__ o, const float* __restrict__ t,
              const float* __restrict__ flags, float* __restrict__ part) {
  __shared__ fvec4 lds[2][2048];   // [buf][idx4]: 0..1023 = output tile, 1024.. = target tile
  const int tid = threadIdx.x;
  const float so = flags[0], fo = flags[1], st = flags[2], ft = flags[3];

  const fvec4* o4 = (const fvec4*)o;
  const fvec4* t4 = (const fvec4*)t;

  float sLH = 0.f, sHL = 0.f, sHH = 0.f, sLL = 0.f;
  float sLH2 = 0.f, sHL2 = 0.f, sHH2 = 0.f;

  // Issue one tile's loads: 8 async b128 instructions per thread.
  auto prefetch = [&](int buf, int tl) {
    const int img   = tl >> 6;            // 0..95
    const int trow  = tl & 63;            // 8-row band within image
    const int base4 = img * 65536 + trow * 1024;  // float4 index of tile start
#pragma unroll
    for (int j = 0; j < 4; ++j) {
      const int idx4 = tid + 256 * j;     // 0..1023 (row*128 + colgroup)
      const unsigned ldsO = (unsigned)((buf * 2048 + idx4) * 16);
      const unsigned ldsT = (unsigned)((buf * 2048 + 1024 + idx4) * 16);
      const fvec4* go = o4 + base4 + idx4;
      const fvec4* gt = t4 + base4 + idx4;
      asm volatile("global_load_async_to_lds_b128 %0, %1, off"
                   :: "v"(ldsO), "v"(go) : "memory");
      asm volatile("global_load_async_to_lds_b128 %0, %1, off"
                   :: "v"(ldsT), "v"(gt) : "memory");
    }
  };

  int tile = blockIdx.x;
  prefetch(0, tile);
  int cur = 0;

#pragma unroll 1
  for (int k = 0; k < TILES_PER_BLOCK; ++k) {
    if (k + 1 < TILES_PER_BLOCK) {
      prefetch(cur ^ 1, tile + NB2);
      // 8 ops outstanding for next tile; <=8 means current tile's 8 are done.
      asm volatile("s_wait_asynccnt 0x8" ::: "memory");
    } else {
      asm volatile("s_wait_asynccnt 0x0" ::: "memory");
    }
    __syncthreads();   // all waves' async writes for buffer `cur` are visible

    // Each thread owns a 4x4 block: rows 4*ty..+3, cols 4*tx..+3 of the tile.
    const int tx = tid & 127;
    const int ty = tid >> 7;
    float d[4][4];
#pragma unroll
    for (int r = 0; r < 4; ++r) {
      const int row = ty * 4 + r;
      fvec4 ov = lds[cur][row * 128 + tx];
      fvec4 tv = lds[cur][1024 + row * 128 + tx];
      d[r][0] = (so * ov.x + fo) - (st * tv.x + ft);
      d[r][1] = (so * ov.y + fo) - (st * tv.y + ft);
      d[r][2] = (so * ov.z + fo) - (st * tv.z + ft);
      d[r][3] = (so * ov.w + fo) - (st * tv.w + ft);
    }
    // Level-1 Haar on the 4 quads of the difference field.
    float qll[2][2];
#pragma unroll
    for (int qr = 0; qr < 2; ++qr)
#pragma unroll
      for (int qc = 0; qc < 2; ++qc) {
        const float a = d[2 * qr][2 * qc],     b = d[2 * qr][2 * qc + 1];
        const float c = d[2 * qr + 1][2 * qc], e = d[2 * qr + 1][2 * qc + 1];
        const float ll = 0.5f * ( a + b + c + e);
        const float lh = 0.5f * (-a - b + c + e);
        const float hl = 0.5f * (-a + b - c + e);
        const float hh = 0.5f * ( a - b - c + e);
        sLL += fabsf(ll); sLH += fabsf(lh); sHL += fabsf(hl); sHH += fabsf(hh);
        qll[qr][qc] = ll;
      }
    // Level-2 Haar on the LL-difference quad (transform is linear).
    {
      const float a = qll[0][0], b = qll[0][1], c = qll[1][0], e = qll[1][1];
      sLH2 += fabsf(0.5f * (-a - b + c + e));
      sHL2 += fabsf(0.5f * (-a + b - c + e));
      sHH2 += fabsf(0.5f * ( a - b - c + e));
    }
    __syncthreads();   // done reading buffer `cur` before it gets overwritten
    cur ^= 1;
    tile += NB2;
  }

  // Deterministic block reduction of 7 channels (LDS reused after final sync).
  float* red = (float*)&lds[0][0];
  red[0 * 256 + tid] = sLH;  red[1 * 256 + tid] = sHL;
  red[2 * 256 + tid] = sHH;  red[3 * 256 + tid] = sLL;
  red[4 * 256 + tid] = sLH2; red[5 * 256 + tid] = sHL2;
  red[6 * 256 + tid] = sHH2;
  __syncthreads();
  for (int s = 128; s > 0; s >>= 1) {
    if (tid < s) {
#pragma unroll
      for (int ch = 0; ch < 7; ++ch)
        red[ch * 256 + tid] += red[ch * 256 + tid + s];
    }
    __syncthreads();
  }
  if (tid == 0) {
#pragma unroll
    for (int ch = 0; ch < 7; ++ch)
      part[ch * NB2 + blockIdx.x] = red[ch * 256];
  }
}

// ---------------------------------------------------- pass 3: WMMA reduction
// 7 waves, one per loss channel. V_WMMA_F32_16X16X4_F32 with B=ones(4x16)
// accumulates row-sums of A into C; with A carrying 64 partials per call the
// grand total is sum_m C[m][0] (lane0 has m=0..7, lane16 has m=8..15).
__global__ __launch_bounds__(224)
void final_pass(const float* __restrict__ part, float* __restrict__ out) {
  __shared__ float chsum[7];
  const int wave = threadIdx.x >> 5;   // 0..6
  const int lane = threadIdx.x & 31;
  const float* p = part + wave * NB2;

  v8f c = {0.f, 0.f, 0.f, 0.f, 0.f, 0.f, 0.f, 0.f};
  v2f ones; ones.x = 1.0f; ones.y = 1.0f;
#pragma unroll
  for (int it = 0; it < NB2 / 64; ++it) {   // 12 WMMA calls, 64 values each
    v2f a;
    a.x = p[it * 64 + lane];
    a.y = p[it * 64 + 32 + lane];
    c = __builtin_amdgcn_wmma_f32_16x16x4_f32(
        /*neg_a=*/false, a, /*neg_b=*/false, ones,
        /*c_mod=*/(short)0, c, /*reuse_a=*/false, /*reuse_b=*/false);
  }
  float s = c[0] + c[1] + c[2] + c[3] + c[4] + c[5] + c[6] + c[7];
  s += __shfl_xor(s, 16, 32);          // combine M=0..7 (lane0) with M=8..15 (lane16)
  if (lane == 0) chsum[wave] = s;
  __syncthreads();
  if (threadIdx.x == 0) {
    const float inv1 = 1.0f / 6291456.0f;   // 32*3*256*256
    const float inv2 = 1.0f / 1572864.0f;   // 32*3*128*128
    out[0] = (chsum[0] + chsum[1] + chsum[2]) * inv1
           + 0.1f * (chsum[3] * inv1)
           + 0.5f * ((chsum[4] + chsum[5] + chsum[6]) * inv2);
  }
}

// ---------------------------------------------------------------- launcher
extern "C" void kernel_launch(void* const* d_in, const int* in_sizes, int n_in,
                              void* d_out, int out_size, void* d_ws, size_t ws_size,
                              hipStream_t stream) {
  const float* o = (const float*)d_in[0];
  const float* t = (const float*)d_in[1];
  float* ws    = (float*)d_ws;
  float* flags = ws;                 // 4 floats
  float* minO  = ws + 16;            // 1024
  float* minT  = ws + 16 + NB1;      // 1024
  float* part  = ws + 16 + 2 * NB1;  // 7*768 floats

  min_pass  <<<NB1, 256, 0, stream>>>(o, t, minO, minT);
  flags_pass<<<1,   256, 0, stream>>>(minO, minT, flags);
  dwt_pass  <<<NB2, 256, 0, stream>>>(o, t, flags, part);
  final_pass<<<1,   224, 0, stream>>>(part, (float*)d_out);
}